// CenterLoss_15925738733576
// MI455X (gfx1250) — compile-verified
//
#include <hip/hip_runtime.h>
#include <hip/hip_bf16.h>

// CenterLoss on MI455X (gfx1250, wave32).
//
// Key insight: the reference's [N,C] distmat is gathered at column labels[i],
// so only the diagonal terms ||x_i - centers[labels[i]]||^2 are needed.
// This kernel is HBM-bandwidth bound (~34 MB traffic -> ~1.5us at 23.3 TB/s).
//
// Per 16-row tile (one wave32):
//   1) stage diff^2 = (x - c_label)^2 into LDS with fully coalesced float4
//      global loads (stride-132 LDS rows -> bank-conflict-free fragment reads)
//   2) reduce rows over D=128 on the matrix pipe: 32 chained
//      V_WMMA_F32_16X16X4_F32 ops with B = ones (every D column = row sum)
//   3) clip per row, sum the 16 rows, emit one partial per tile
// A deterministic single-block reduction produces the mean.

typedef __attribute__((ext_vector_type(2))) float v2f;
typedef __attribute__((ext_vector_type(4))) float v4f;
typedef __attribute__((ext_vector_type(8))) float v8f;

#define WAVES_PER_BLOCK 8
#define TILE_ROWS 16
#define D_DIM 128
#define LDS_STRIDE 132  // 128 + 4 pad: (132*m + k) % 64 banks distinct over fragment

__global__ __launch_bounds__(256) void center_loss_tile_kernel(
    const float* __restrict__ x,
    const int* __restrict__ labels,
    const float* __restrict__ centers,
    float* __restrict__ partial,
    int ntiles)
{
    __shared__ float diff2[WAVES_PER_BLOCK][TILE_ROWS * LDS_STRIDE];

    const int tid  = (int)threadIdx.x;
    const int lane = tid & 31;
    const int wave = tid >> 5;
    const int tile = (int)blockIdx.x * WAVES_PER_BLOCK + wave;
    // Clamp (instead of early-return) so every thread reaches __syncthreads()
    // and EXEC stays all-ones for the WMMA ops.
    const int tile_c = (tile < ntiles) ? tile : (ntiles - 1);
    const int row0   = tile_c * TILE_ROWS;

    float* lds = &diff2[wave][0];

    // 16 labels for this tile, one per low lane; broadcast per row via shuffle.
    int lab = labels[row0 + (lane & 15)];

    // ---- Stage (x - c)^2 into LDS: fully coalesced 512B/row global loads ----
#pragma unroll
    for (int t = 0; t < TILE_ROWS; ++t) {
        const int lt = __shfl(lab, t, 32);
        const v4f* __restrict__ xr = (const v4f*)(x + (size_t)(row0 + t) * D_DIM);
        const v4f* __restrict__ cr = (const v4f*)(centers + (size_t)lt * D_DIM);
        v4f d = xr[lane] - cr[lane];
        *(v4f*)(lds + t * LDS_STRIDE + 4 * lane) = d * d;
    }
    __syncthreads();

    // ---- Row-reduce over D via WMMA with ones: D[m][*] = sum_k A[m][k] ----
    // A-fragment (16x4 f32): lanes 0-15 hold row m, K={k0,k0+1};
    //                        lanes 16-31 hold row m, K={k0+2,k0+3}.
    const int m  = lane & 15;
    const int kh = lane >> 4;
    const float* arow = lds + m * LDS_STRIDE + 2 * kh;

    v8f acc = {0.f, 0.f, 0.f, 0.f, 0.f, 0.f, 0.f, 0.f};
    const v2f ones = {1.0f, 1.0f};
#pragma unroll
    for (int k0 = 0; k0 < D_DIM; k0 += 4) {
        v2f a = *(const v2f*)(arow + k0);  // ds_load_b64, 8B aligned, conflict-free
        acc = __builtin_amdgcn_wmma_f32_16x16x4_f32(
            /*neg_a=*/false, a, /*neg_b=*/false, ones,
            /*c_mod=*/(short)0, acc, /*reuse_a=*/false, /*reuse_b=*/false);
    }

    // C/D layout: VGPR r, lanes 0-15 -> row r; lanes 16-31 -> row r+8.
    // All 16 columns identical, so every lane of each half holds the same sums.
    float s = 0.0f;
#pragma unroll
    for (int r = 0; r < 8; ++r) {
        float v = acc[r];
        v = fminf(fmaxf(v, 1e-12f), 1e12f);  // clip before summing (per-row)
        s += v;
    }
    // lanes 0-15: rows 0-7 sum; lanes 16-31: rows 8-15 sum -> combine halves.
    float tot = s + __shfl_xor(s, 16, 32);

    if (lane == 0 && tile < ntiles) partial[tile] = tot;
}

__global__ __launch_bounds__(256) void center_loss_reduce_kernel(
    const float* __restrict__ partial, float* __restrict__ out,
    int n, float inv_n)
{
    __shared__ float sm[256];
    float s = 0.0f;
    for (int i = (int)threadIdx.x; i < n; i += 256) s += partial[i];
    sm[threadIdx.x] = s;
    __syncthreads();
#pragma unroll
    for (int off = 128; off > 0; off >>= 1) {
        if ((int)threadIdx.x < off) sm[threadIdx.x] += sm[threadIdx.x + off];
        __syncthreads();
    }
    if (threadIdx.x == 0) out[0] = sm[0] * inv_n;
}

extern "C" void kernel_launch(void* const* d_in, const int* in_sizes, int n_in,
                              void* d_out, int out_size, void* d_ws, size_t ws_size,
                              hipStream_t stream)
{
    const float* x       = (const float*)d_in[0];   // [N, 128] f32
    const int*   labels  = (const int*)d_in[1];     // [N] int32
    const float* centers = (const float*)d_in[2];   // [C, 128] f32
    float*       out     = (float*)d_out;           // scalar mean
    float*       partial = (float*)d_ws;            // [ntiles] partials

    const int N      = in_sizes[1];
    const int ntiles = (N + TILE_ROWS - 1) / TILE_ROWS;
    const int blocks = (ntiles + WAVES_PER_BLOCK - 1) / WAVES_PER_BLOCK;

    center_loss_tile_kernel<<<blocks, 256, 0, stream>>>(x, labels, centers,
                                                        partial, ntiles);
    center_loss_reduce_kernel<<<1, 256, 0, stream>>>(partial, out, ntiles,
                                                     1.0f / (float)N);
}